// Warp_45062796869769
// MI455X (gfx1250) — compile-verified
//
#include <hip/hip_runtime.h>
#include <stdint.h>

// Problem shape (fixed by the reference's setup_inputs):
//   left_feat  [1, 32, 128, 256] f32  (UNUSED by the reference output)
//   right_feat [1, 32, 128, 256] f32
//   disparity  [1, 48, 128, 256] i32
//   out        [1, 32, 48, 128, 256] f32  (201 MB, streaming, write-once)
#define Ck 32
#define Hk 128
#define Wk 256
#define Dk 48
#define D_PER_BLK 4   // disparities per block (one per 64-thread quarter)

// Native clang vectors (required by __builtin_nontemporal_*; HIP float4/int4
// are structs and are rejected).
typedef __attribute__((ext_vector_type(4))) float fvec4;
typedef __attribute__((ext_vector_type(4))) int   ivec4;

__global__ __launch_bounds__(256) void warp_gather_kernel(
    const float* __restrict__ right,   // [C,H,W]  (4 MB, L2-resident, keep RT)
    const int*   __restrict__ disp,    // [D,H,W]  (6 MB, read once -> NT)
    float*       __restrict__ out)     // [C,D,H,W] (write once -> NT)
{
    // 32 KB LDS slab holding right[:, h, :] as smem[c*Wk + w]
    __shared__ float smem[Ck * Wk];

    const int tid  = threadIdx.x;
    const int h    = (int)blockIdx.x % Hk;
    const int dblk = (int)blockIdx.x / Hk;

    // ---- Stage right[:, h, :] into LDS with CDNA5 async global->LDS DMA ----
    // 8192 floats; 256 threads x 16B x 8 iterations. Each lane issues its own
    // b128 async copy (ASYNCcnt), then we drain with s_wait_asynccnt + barrier.
    const unsigned lds_base = (unsigned)(uintptr_t)(&smem[0]); // low 32b of flat = LDS offset
    #pragma unroll
    for (int i = 0; i < (Ck * Wk) / (256 * 4); ++i) {
        const int e = (i * 256 + tid) * 4;        // float index, 16B aligned
        const int c = e >> 8;                     // e / Wk  (Wk == 256)
        const int w = e & (Wk - 1);
        const unsigned long long gaddr =
            (unsigned long long)(uintptr_t)(right + ((size_t)c * Hk + h) * (size_t)Wk + w);
        const unsigned laddr = lds_base + (unsigned)e * 4u;
        asm volatile("global_load_async_to_lds_b128 %0, %1, off"
                     :: "v"(laddr), "v"(gaddr)
                     : "memory");
    }
    asm volatile("s_wait_asynccnt 0" ::: "memory");
    __syncthreads();

    // ---- Per-thread warp/gather work ----
    // quarter q (0..3) owns disparity d = dblk*4 + q; lane-in-quarter owns 4 w's.
    const int q  = tid >> 6;
    const int w0 = (tid & 63) << 2;     // 0,4,...,252
    const int d  = dblk * D_PER_BLK + q;

    // Coalesced 16B load of 4 disparities; read exactly once -> non-temporal
    const ivec4 dv = __builtin_nontemporal_load(
        (const ivec4*)(disp + ((size_t)d * Hk + h) * (size_t)Wk + w0));

    const int rc0 = w0 + 0 - dv.x;
    const int rc1 = w0 + 1 - dv.y;
    const int rc2 = w0 + 2 - dv.z;
    const int rc3 = w0 + 3 - dv.w;

    // valid iff 0 <= rc <= W-1 (matches reference: right_coord in [0, W-1])
    const float m0 = (rc0 >= 0 && rc0 < Wk) ? 1.0f : 0.0f;
    const float m1 = (rc1 >= 0 && rc1 < Wk) ? 1.0f : 0.0f;
    const float m2 = (rc2 >= 0 && rc2 < Wk) ? 1.0f : 0.0f;
    const float m3 = (rc3 >= 0 && rc3 < Wk) ? 1.0f : 0.0f;

    const int i0 = min(max(rc0, 0), Wk - 1);
    const int i1 = min(max(rc1, 0), Wk - 1);
    const int i2 = min(max(rc2, 0), Wk - 1);
    const int i3 = min(max(rc3, 0), Wk - 1);

    float* obase = out + ((size_t)d * Hk + h) * (size_t)Wk + w0;
    const size_t c_stride = (size_t)Dk * Hk * Wk;   // out channel stride

    #pragma unroll 8
    for (int c = 0; c < Ck; ++c) {
        const float* sr = &smem[c * Wk];  // LDS gathers (ds_load_2addr_b32 pairs)
        fvec4 v;
        v.x = sr[i0] * m0;
        v.y = sr[i1] * m1;
        v.z = sr[i2] * m2;
        v.w = sr[i3] * m3;
        // Streaming 16B store, 512B contiguous per wave32, non-temporal so the
        // 201 MB output doesn't wash the L2 that keeps right_feat resident.
        __builtin_nontemporal_store(v, (fvec4*)(obase + (size_t)c * c_stride));
    }
}

extern "C" void kernel_launch(void* const* d_in, const int* in_sizes, int n_in,
                              void* d_out, int out_size, void* d_ws, size_t ws_size,
                              hipStream_t stream) {
    (void)in_sizes; (void)n_in; (void)d_ws; (void)ws_size; (void)out_size;
    // d_in[0] = left_feat (unused by the reference output)
    const float* right = (const float*)d_in[1];
    const int*   disp  = (const int*)d_in[2];
    float*       out   = (float*)d_out;

    dim3 grid(Hk * (Dk / D_PER_BLK));   // 128 * 12 = 1536 blocks
    warp_gather_kernel<<<grid, 256, 0, stream>>>(right, disp, out);
}